// SAGE_3393024164201
// MI455X (gfx1250) — compile-verified
//
#include <hip/hip_runtime.h>

typedef __attribute__((ext_vector_type(2))) float v2f;
typedef __attribute__((ext_vector_type(4))) float v4f;
typedef __attribute__((ext_vector_type(8))) float v8f;

#define N_NODES 40000
#define N_EDGES 640000
#define DIM     128
#define BATCH   131072
#define LDSW1   132   // padded LDS row stride (floats) for 128-wide weights
#define LDSW2   260   // padded LDS row stride (floats) for 256-wide weights

#define WMMA_F32(a, b, c) __builtin_amdgcn_wmma_f32_16x16x4_f32( \
    false, (a), false, (b), (short)0, (c), false, false)

// ---------------------------------------------------------------------------
// Kernel 0: zero accumulator + counts
// ---------------------------------------------------------------------------
__global__ void sage_zero_kernel(float* __restrict__ p, int n) {
    int i = blockIdx.x * blockDim.x + threadIdx.x;
    if (i < n) p[i] = 0.0f;
}

// ---------------------------------------------------------------------------
// Kernel 1: scatter-sum edge_attr into accum[dst] + counts.
// One wave per edge; lane l owns floats 4l..4l+3. NT loads: edge_attr is
// read-once (328 MB) and must not evict the L2-resident accumulator.
// ---------------------------------------------------------------------------
__global__ void sage_scatter_kernel(const float* __restrict__ edge_attr,
                                    const int*   __restrict__ dst,
                                    float* __restrict__ accum,
                                    float* __restrict__ cnt) {
    int lane = threadIdx.x & 31;
    int edge = blockIdx.x * (blockDim.x >> 5) + (threadIdx.x >> 5);
    if (edge >= N_EDGES) return;
    int d = dst[edge];
    v4f v = __builtin_nontemporal_load(
        (const v4f*)(edge_attr + (size_t)edge * DIM + lane * 4));
    float* base = accum + (size_t)d * DIM + lane * 4;
    atomicAdd(base + 0, v.x);
    atomicAdd(base + 1, v.y);
    atomicAdd(base + 2, v.z);
    atomicAdd(base + 3, v.w);
    if (lane == 0) atomicAdd(cnt + d, 1.0f);
}

// ---------------------------------------------------------------------------
// Kernel 2: h = relu( (accum / max(cnt,1)) @ Wn^T + bn )
// 256 thr = 8 waves / block; wave owns 32 rows x 128 cols (2 M-tiles).
// Wn staged in LDS (stride 132 -> conflict-free ds_load_b64 across 64 banks).
// Double-buffered K loop: next-step A/B loads issued before current WMMAs.
// ---------------------------------------------------------------------------
__global__ __launch_bounds__(256) void sage_node_gemm_kernel(
    const float* __restrict__ accum, const float* __restrict__ cnt,
    const float* __restrict__ Wn, const float* __restrict__ bnb,
    float* __restrict__ h)
{
    __shared__ __align__(16) float bsh[DIM * LDSW1];
    for (int idx = threadIdx.x; idx < DIM * (DIM / 4); idx += 256) {
        int r = idx >> 5;            // weight output row n (0..127)
        int c = (idx & 31) * 4;      // k
        *(v4f*)(bsh + r * LDSW1 + c) = *(const v4f*)(Wn + r * DIM + c);
    }
    __syncthreads();

    int lane = threadIdx.x & 31;
    int wave = blockIdx.x * 8 + (threadIdx.x >> 5);
    if (wave >= N_NODES / 32) return;          // wave-uniform: EXEC stays all-1

    const int ml = lane & 15;
    const int kh = (lane >> 4) * 2;
    const int r0 = wave * 32;
    const int rowA = r0 + ml, rowB = r0 + 16 + ml;
    const float inv0 = 1.0f / fmaxf(cnt[rowA], 1.0f);
    const float inv1 = 1.0f / fmaxf(cnt[rowB], 1.0f);
    const float* a0p = accum + (size_t)rowA * DIM + kh;
    const float* a1p = accum + (size_t)rowB * DIM + kh;
    const float* bp  = bsh + ml * LDSW1 + kh;  // + t*16*LDSW1 + k0

    v8f acc0[8], acc1[8];
#pragma unroll
    for (int t = 0; t < 8; ++t) {
        acc0[t] = v8f{0.f,0.f,0.f,0.f,0.f,0.f,0.f,0.f};
        acc1[t] = v8f{0.f,0.f,0.f,0.f,0.f,0.f,0.f,0.f};
    }

    // prologue
    v2f a0 = *(const v2f*)a0p;  a0.x *= inv0; a0.y *= inv0;
    v2f a1 = *(const v2f*)a1p;  a1.x *= inv1; a1.y *= inv1;
    v2f bc[8];
#pragma unroll
    for (int t = 0; t < 8; ++t) bc[t] = *(const v2f*)(bp + t * 16 * LDSW1);

#pragma unroll 4
    for (int k0 = 0; k0 < DIM - 4; k0 += 4) {
        // issue next-step loads first
        v2f a0n = *(const v2f*)(a0p + k0 + 4);
        v2f a1n = *(const v2f*)(a1p + k0 + 4);
        v2f bn[8];
#pragma unroll
        for (int t = 0; t < 8; ++t)
            bn[t] = *(const v2f*)(bp + t * 16 * LDSW1 + k0 + 4);
        a0n.x *= inv0; a0n.y *= inv0;
        a1n.x *= inv1; a1n.y *= inv1;
        // consume current buffers
#pragma unroll
        for (int t = 0; t < 8; ++t) {
            acc0[t] = WMMA_F32(a0, bc[t], acc0[t]);
            acc1[t] = WMMA_F32(a1, bc[t], acc1[t]);
        }
        a0 = a0n; a1 = a1n;
#pragma unroll
        for (int t = 0; t < 8; ++t) bc[t] = bn[t];
    }
    // drain
#pragma unroll
    for (int t = 0; t < 8; ++t) {
        acc0[t] = WMMA_F32(a0, bc[t], acc0[t]);
        acc1[t] = WMMA_F32(a1, bc[t], acc1[t]);
    }

    // epilogue: d vgpr j -> row = base + j + 8*(lane>>4), col = t*16 + ml
    const int mb0 = r0 + (lane >> 4) * 8;
    const int mb1 = r0 + 16 + (lane >> 4) * 8;
#pragma unroll
    for (int t = 0; t < 8; ++t) {
        const int n = t * 16 + ml;
        const float bias = bnb[n];
#pragma unroll
        for (int j = 0; j < 8; ++j) {
            h[(size_t)(mb0 + j) * DIM + n] = fmaxf(acc0[t][j] + bias, 0.0f);
            h[(size_t)(mb1 + j) * DIM + n] = fmaxf(acc1[t][j] + bias, 0.0f);
        }
    }
}

// ---------------------------------------------------------------------------
// Kernel 3: e = h[src] @ We[:, :128]^T + h[dst] @ We[:, 128:]^T + be
// Wave owns 32 batch rows x 128 cols; We (128x256) staged in LDS (stride 260).
// Two pipelined K=128 passes (src-half then dst-half). NT stores for e.
// ---------------------------------------------------------------------------
__global__ __launch_bounds__(256) void sage_edge_gemm_kernel(
    const float* __restrict__ h, const int* __restrict__ couples,
    const float* __restrict__ We, const float* __restrict__ be,
    float* __restrict__ e)
{
    __shared__ __align__(16) float bsh[DIM * LDSW2];
    for (int idx = threadIdx.x; idx < DIM * (2 * DIM / 4); idx += 256) {
        int r = idx >> 6;            // weight output row n (0..127)
        int c = (idx & 63) * 4;      // k (0..255)
        *(v4f*)(bsh + r * LDSW2 + c) = *(const v4f*)(We + r * (2 * DIM) + c);
    }
    __syncthreads();

    int lane = threadIdx.x & 31;
    int wave = blockIdx.x * 8 + (threadIdx.x >> 5);
    if (wave >= BATCH / 32) return;

    const int ml = lane & 15;
    const int kh = (lane >> 4) * 2;
    const int r0 = wave * 32;
    const int rA = r0 + ml, rB = r0 + 16 + ml;
    const int srcA = couples[rA * 2 + 0], dstA = couples[rA * 2 + 1];
    const int srcB = couples[rB * 2 + 0], dstB = couples[rB * 2 + 1];
    const float* bp = bsh + ml * LDSW2 + kh;

    v8f acc0[8], acc1[8];
#pragma unroll
    for (int t = 0; t < 8; ++t) {
        acc0[t] = v8f{0.f,0.f,0.f,0.f,0.f,0.f,0.f,0.f};
        acc1[t] = v8f{0.f,0.f,0.f,0.f,0.f,0.f,0.f,0.f};
    }

#pragma unroll
    for (int pass = 0; pass < 2; ++pass) {
        const float* a0p = h + (size_t)(pass ? dstA : srcA) * DIM + kh;
        const float* a1p = h + (size_t)(pass ? dstB : srcB) * DIM + kh;
        const float* bpp = bp + pass * DIM;     // cols 0..127 / 128..255

        v2f a0 = *(const v2f*)a0p;
        v2f a1 = *(const v2f*)a1p;
        v2f bc[8];
#pragma unroll
        for (int t = 0; t < 8; ++t) bc[t] = *(const v2f*)(bpp + t * 16 * LDSW2);

#pragma unroll 4
        for (int k0 = 0; k0 < DIM - 4; k0 += 4) {
            v2f a0n = *(const v2f*)(a0p + k0 + 4);
            v2f a1n = *(const v2f*)(a1p + k0 + 4);
            v2f bn[8];
#pragma unroll
            for (int t = 0; t < 8; ++t)
                bn[t] = *(const v2f*)(bpp + t * 16 * LDSW2 + k0 + 4);
#pragma unroll
            for (int t = 0; t < 8; ++t) {
                acc0[t] = WMMA_F32(a0, bc[t], acc0[t]);
                acc1[t] = WMMA_F32(a1, bc[t], acc1[t]);
            }
            a0 = a0n; a1 = a1n;
#pragma unroll
            for (int t = 0; t < 8; ++t) bc[t] = bn[t];
        }
#pragma unroll
        for (int t = 0; t < 8; ++t) {
            acc0[t] = WMMA_F32(a0, bc[t], acc0[t]);
            acc1[t] = WMMA_F32(a1, bc[t], acc1[t]);
        }
    }

    const int mb0 = r0 + (lane >> 4) * 8;
    const int mb1 = r0 + 16 + (lane >> 4) * 8;
#pragma unroll
    for (int t = 0; t < 8; ++t) {
        const int n = t * 16 + ml;
        const float bias = be[n];
#pragma unroll
        for (int j = 0; j < 8; ++j) {
            __builtin_nontemporal_store(acc0[t][j] + bias,
                                        e + (size_t)(mb0 + j) * DIM + n);
            __builtin_nontemporal_store(acc1[t][j] + bias,
                                        e + (size_t)(mb1 + j) * DIM + n);
        }
    }
}

// ---------------------------------------------------------------------------
// Launch. Inputs: 0:x 1:edge_index 2:edge_attr 3:edge_couples 4..7:layer-1
// (dead) 8:Wn2 9:bn2 10:We2 11:be2. Output = layer-2 e only.
// Workspace: [accum 40000*128][cnt 40000][h 40000*128] f32 (~41 MB).
// ---------------------------------------------------------------------------
extern "C" void kernel_launch(void* const* d_in, const int* in_sizes, int n_in,
                              void* d_out, int out_size, void* d_ws, size_t ws_size,
                              hipStream_t stream) {
    const int*   edge_index = (const int*)d_in[1];
    const float* edge_attr  = (const float*)d_in[2];
    const int*   couples    = (const int*)d_in[3];
    const float* Wn2        = (const float*)d_in[8];
    const float* bn2        = (const float*)d_in[9];
    const float* We2        = (const float*)d_in[10];
    const float* be2        = (const float*)d_in[11];
    float*       out        = (float*)d_out;

    const int* dst = edge_index + N_EDGES;     // edge_index[1]

    float* accum = (float*)d_ws;
    float* cnt   = accum + (size_t)N_NODES * DIM;
    float* h     = cnt + N_NODES;

    {
        int n = N_NODES * DIM + N_NODES;
        sage_zero_kernel<<<(n + 255) / 256, 256, 0, stream>>>(accum, n);
    }
    sage_scatter_kernel<<<N_EDGES / 8, 256, 0, stream>>>(edge_attr, dst, accum, cnt);
    sage_node_gemm_kernel<<<(N_NODES / 32 + 7) / 8, 256, 0, stream>>>(accum, cnt, Wn2, bn2, h);
    sage_edge_gemm_kernel<<<BATCH / 32 / 8, 256, 0, stream>>>(h, couples, We2, be2, out);
}